// RGCNLayer_38345468019172
// MI455X (gfx1250) — compile-verified
//
#include <hip/hip_runtime.h>

typedef __attribute__((ext_vector_type(2))) float v2f;
typedef __attribute__((ext_vector_type(8))) float v8f;

#define IN_FEATS  128
#define OUT_FEATS 128
#define N_RELS    8

// ---------------------------------------------------------------------------
// Kernel 1: zero the output accumulator (float4 stores)
// ---------------------------------------------------------------------------
__global__ __launch_bounds__(256) void rgcn_zero_f4(float4* __restrict__ p, int n4) {
    int i = blockIdx.x * 256 + threadIdx.x;
    if (i < n4) p[i] = make_float4(0.f, 0.f, 0.f, 0.f);
}

// ---------------------------------------------------------------------------
// Kernel 2: ht[n, r, :] = h[n, :] @ W[r]   via V_WMMA_F32_16X16X4_F32
//
// One block (256 thr = 8 waves) per 16-node tile. The 16x128 h-tile is staged
// in LDS once (coalesced). Wave w owns output columns [16w, 16w+16); it loops
// over the 8 relations, doing 32 K-steps of 16x16x4 f32 WMMA per relation.
//
// A operand (16x4 f32, 2 VGPRs): lanes 0-15 hold (M=lane, K=k+0/k+1),
//                                lanes 16-31 hold (M=lane-16, K=k+2/k+3).
// B operand (4x16 f32, 2 VGPRs): lanes 0-15 hold (K=k+0/k+1, N=lane),
//                                lanes 16-31 hold (K=k+2/k+3, N=lane-16).
// D (16x16 f32, 8 VGPRs): VGPR v -> M = v (lanes 0-15) / v+8 (lanes 16-31),
//                         N = lane & 15.
// ---------------------------------------------------------------------------
__global__ __launch_bounds__(256) void rgcn_transform(
    const float* __restrict__ h,       // [N, 128]
    const float* __restrict__ W,       // [8, 128, 128]
    float* __restrict__ ht,            // [N, 8, 128]
    int n_nodes)
{
    __shared__ float hs[16 * IN_FEATS];          // 8 KB h tile

    const int tid       = threadIdx.x;
    const int lane      = tid & 31;
    const int wave      = tid >> 5;
    const int node_base = blockIdx.x * 16;
    if (node_base >= n_nodes) return;

    // Cooperative coalesced load of the 16x128 tile (2048 floats = 512 float4)
    {
        const float4* src = (const float4*)(h + (size_t)node_base * IN_FEATS);
        float4*       dst = (float4*)hs;
        dst[tid]       = src[tid];
        dst[tid + 256] = src[tid + 256];
    }
    __syncthreads();

    const int ob   = wave * 16;        // this wave's output-column tile
    const int half = lane >> 4;        // 0: K pair {0,1}, 1: K pair {2,3}
    const int mn   = lane & 15;        // M index (for A) / N index (for B,D)

    for (int r = 0; r < N_RELS; ++r) {
        const float* Wr = W + (size_t)r * IN_FEATS * OUT_FEATS;
        v8f acc = {};

        #pragma unroll 4
        for (int k = 0; k < IN_FEATS; k += 4) {
            const int ka = k + half * 2;
            // A from LDS: two consecutive floats -> one ds_load_b64
            v2f a = *(const v2f*)(&hs[mn * IN_FEATS + ka]);
            // B from global (W row-major, lanes coalesced across N)
            v2f b;
            b.x = Wr[(size_t)ka       * OUT_FEATS + ob + mn];
            b.y = Wr[(size_t)(ka + 1) * OUT_FEATS + ob + mn];
            acc = __builtin_amdgcn_wmma_f32_16x16x4_f32(
                /*neg_a=*/false, a, /*neg_b=*/false, b,
                /*c_mod=*/(short)0, acc, /*reuse_a=*/false, /*reuse_b=*/false);
        }

        // Scatter D tile: element (M, N) -> ht[(node_base+M)*8 + r][ob + N]
        float* outp = ht + (((size_t)node_base * N_RELS + r) * OUT_FEATS) + ob + mn;
        #pragma unroll
        for (int v = 0; v < 8; ++v) {
            const int M = v + half * 8;
            outp[(size_t)M * N_RELS * OUT_FEATS] = acc[v];
        }
    }
}

// ---------------------------------------------------------------------------
// Kernel 3: per-edge gather + scale + atomic scatter-sum.
// One wave per edge: float4 gather of the 512B ht row, 4 f32 atomic adds.
// ---------------------------------------------------------------------------
__global__ __launch_bounds__(256) void rgcn_scatter(
    const float* __restrict__ ht,      // [N, 8, 128]
    const int*   __restrict__ src,
    const int*   __restrict__ dst,
    const int*   __restrict__ etype,
    const float* __restrict__ norm,
    float* __restrict__ out,           // [N, 128]
    int n_edges)
{
    const int e    = (blockIdx.x * 256 + threadIdx.x) >> 5;
    const int lane = threadIdx.x & 31;
    if (e >= n_edges) return;

    const int   s = src[e];
    const int   d = dst[e];
    const int   t = etype[e];
    const float w = norm[e];

    const float4* row = (const float4*)(ht + ((size_t)s * N_RELS + t) * OUT_FEATS);
    float4 v = row[lane];

    float* o = out + (size_t)d * OUT_FEATS + lane * 4;
    unsafeAtomicAdd(o + 0, v.x * w);   // global_atomic_add_f32 (no return)
    unsafeAtomicAdd(o + 1, v.y * w);
    unsafeAtomicAdd(o + 2, v.z * w);
    unsafeAtomicAdd(o + 3, v.w * w);
}

// ---------------------------------------------------------------------------
// Kernel 4: out = relu(out + bias)
// ---------------------------------------------------------------------------
__global__ __launch_bounds__(256) void rgcn_bias_relu(
    float* __restrict__ out, const float* __restrict__ bias, int n)
{
    int i = blockIdx.x * 256 + threadIdx.x;
    if (i < n) {
        float v = out[i] + bias[i & (OUT_FEATS - 1)];
        out[i] = v > 0.f ? v : 0.f;
    }
}

// ---------------------------------------------------------------------------
extern "C" void kernel_launch(void* const* d_in, const int* in_sizes, int n_in,
                              void* d_out, int out_size, void* d_ws, size_t ws_size,
                              hipStream_t stream) {
    const float* h         = (const float*)d_in[0];
    const float* edge_norm = (const float*)d_in[1];
    const int*   edge_type = (const int*)d_in[2];
    const int*   src       = (const int*)d_in[3];
    const int*   dst       = (const int*)d_in[4];
    const float* weight    = (const float*)d_in[5];
    const float* bias      = (const float*)d_in[6];
    float*       out       = (float*)d_out;
    float*       ht        = (float*)d_ws;   // [n_nodes, 8, 128] = 204.8 MB

    const int n_nodes = in_sizes[0] / IN_FEATS;
    const int n_edges = in_sizes[1];

    // 1) zero output accumulator
    {
        int n4 = out_size / 4;
        rgcn_zero_f4<<<(n4 + 255) / 256, 256, 0, stream>>>((float4*)out, n4);
    }
    // 2) per-relation transform (WMMA)
    {
        int blocks = (n_nodes + 15) / 16;
        rgcn_transform<<<blocks, 256, 0, stream>>>(h, weight, ht, n_nodes);
    }
    // 3) gather + atomic scatter-sum (one wave per edge)
    {
        int blocks = (n_edges + 7) / 8;   // 8 waves per 256-thread block
        rgcn_scatter<<<blocks, 256, 0, stream>>>(ht, src, dst, edge_type,
                                                 edge_norm, out, n_edges);
    }
    // 4) bias + ReLU
    {
        rgcn_bias_relu<<<(out_size + 255) / 256, 256, 0, stream>>>(out, bias, out_size);
    }
}